// SAFL_40742059770114
// MI455X (gfx1250) — compile-verified
//
#include <hip/hip_runtime.h>
#include <cstdint>
#include <cstddef>

typedef __attribute__((ext_vector_type(2))) float v2f;
typedef __attribute__((ext_vector_type(8))) float v8f;

constexpr int B = 256;
constexpr int C = 2048;
constexpr int H = 18;
constexpr int W = 9;
constexpr int N = H * W;            // 162
constexpr int P = 6;
constexpr int MPAD = 16;            // pad P -> 16 rows for WMMA
constexpr int NT = 176;             // pad N -> 11 tiles of 16
constexpr int NTILES = NT / 16;     // 11
constexpr int WAVES = NTILES;       // 11 waves per workgroup
constexpr int THREADS = WAVES * 32; // 352
constexpr int CTILES = C / 16;      // 128

__device__ __forceinline__ v2f ld2(const float* p) {
  const float2 t = *reinterpret_cast<const float2*>(p);
  v2f r; r.x = t.x; r.y = t.y;
  return r;
}

__device__ __forceinline__ v8f wmma_f32(v2f a, v2f b, v8f c) {
  return __builtin_amdgcn_wmma_f32_16x16x4_f32(false, a, false, b,
                                               (short)0, c, false, false);
}

// ---------------------------------------------------------------------------
// Kernel 0: posdot[p][n] = sum_c part[p][c] * pos[n][c]   (batch-invariant)
// Stored zero-padded as [16][176] so the main kernel needs no edge logic.
// ---------------------------------------------------------------------------
__global__ void posdot_kernel(const float* __restrict__ part,
                              const float* __restrict__ pos,
                              float* __restrict__ posdot) {
  const int n = blockIdx.x;   // 0..NT-1 (>=N columns produce zero padding)
  const int tid = threadIdx.x;
  __shared__ float red[P][256];
  float acc[P];
#pragma unroll
  for (int p = 0; p < P; ++p) acc[p] = 0.f;
  if (n < N) {
    for (int c = tid; c < C; c += 256) {
      const float xv = pos[n * C + c];
#pragma unroll
      for (int p = 0; p < P; ++p) acc[p] += part[p * C + c] * xv;
    }
  }
#pragma unroll
  for (int p = 0; p < P; ++p) red[p][tid] = acc[p];
  __syncthreads();
  for (int s = 128; s > 0; s >>= 1) {
    if (tid < s) {
#pragma unroll
      for (int p = 0; p < P; ++p) red[p][tid] += red[p][tid + s];
    }
    __syncthreads();
  }
  if (tid < MPAD) {
    float v = 0.f;
    if (tid < P && n < N) v = red[tid][0];
    posdot[tid * NT + n] = v;   // rows 6..15 and cols 162..175 are zero
  }
}

// ---------------------------------------------------------------------------
// Main kernel: one workgroup per batch element (x_b = 1.33 MB, phase-2 re-read
// hits L2). All inner-loop loads are unconditional: out-of-range rows/columns
// only feed discarded D rows/cols (addresses clamped in-bounds), and phase-2
// K-padding is neutralized by zeroed attnLds columns.
// ---------------------------------------------------------------------------
__global__ void __launch_bounds__(THREADS)
safl_main_kernel(const float* __restrict__ x,
                 const float* __restrict__ part,
                 const float* __restrict__ posdot,
                 float* __restrict__ out,
                 float* __restrict__ attn_out) {
  __shared__ __align__(16) float partLds[P * C];      // 48 KB
  __shared__ __align__(16) float attnLds[MPAD * NT];  // 11 KB

  const int b    = blockIdx.x;
  const int tid  = threadIdx.x;
  const int wave = tid >> 5;
  const int lane = tid & 31;
  const int lm   = lane & 15;             // row/col within 16
  const int kr   = (lane >> 4) << 1;      // 0 (lanes 0-15) or 2 (lanes 16-31)
  const int mbase = (lane < 16) ? 0 : 8;  // C/D fragment row base

  // Cooperative load of part tokens into LDS (b128 loads).
  {
    const float4* src = reinterpret_cast<const float4*>(part);
    float4* dst = reinterpret_cast<float4*>(partLds);
    for (int i = tid; i < (P * C) / 4; i += THREADS) dst[i] = src[i];
  }
  __syncthreads();

  const float* xb = x + (size_t)b * C * N;

  // -------------------- Phase 1: attention logits + sigmoid ---------------
  {
    const int n0 = wave * 16;
    const int n  = n0 + lm;
    const bool nValid = (n < N);
    const int nc  = nValid ? n : (N - 1);   // clamp: cols >=162 are discarded
    const int lmc = (lm < P) ? lm : 0;      // clamp: rows >=6 are discarded
    const float* aRow = partLds + lmc * C + kr;
    const float* bCol = xb + nc;
    v8f acc = {};
    for (int k = 0; k < C; k += 16) {       // 4 WMMA steps per group
      const float* p = bCol + (size_t)(k + kr) * N;
      // grouped loads (clause-friendly), then 4 back-to-back WMMAs
      const v2f a0 = ld2(aRow + k);
      const v2f a1 = ld2(aRow + k + 4);
      const v2f a2 = ld2(aRow + k + 8);
      const v2f a3 = ld2(aRow + k + 12);
      v2f b0; b0.x = p[0];      b0.y = p[N];
      v2f b1; b1.x = p[4 * N];  b1.y = p[5 * N];
      v2f b2; b2.x = p[8 * N];  b2.y = p[9 * N];
      v2f b3; b3.x = p[12 * N]; b3.y = p[13 * N];
      // Prefetch one full unrolled iteration (64 K-rows) ahead so the GL2
      // fill leads the demand loads of the *next* asm iteration.
      if (k + 64 < C) __builtin_prefetch(p + (size_t)64 * N, 0, 0);
      acc = wmma_f32(a0, b0, acc);
      acc = wmma_f32(a1, b1, acc);
      acc = wmma_f32(a2, b2, acc);
      acc = wmma_f32(a3, b3, acc);
    }
#pragma unroll
    for (int i = 0; i < 8; ++i) {
      const int M = mbase + i;
      const float v = acc[i] + posdot[M * NT + n0 + lm];
      float s = 1.f / (1.f + __expf(-v));
      if (!nValid) s = 0.f;                 // zero K-padding for phase 2
      attnLds[M * NT + n0 + lm] = s;
      if (nValid && M < P)
        attn_out[((size_t)b * P + M) * N + n] = s;
    }
  }
  __syncthreads();

  // -------------------- Phase 2: out = attn @ x^T / (H*W) -----------------
  const float invN = 1.0f / (float)N;
  for (int ct = wave; ct < CTILES; ct += WAVES) {
    const int c0 = ct * 16;
    const int c  = c0 + lm;
    const float* bRow = xb + (size_t)c * N;   // x_b[c][*], contiguous in n
    const float* aRow = attnLds + lm * NT + kr;
    v8f acc = {};
    for (int k = 0; k < 160; k += 16) {       // K = 0..159, fully in-range
      const v2f a0 = ld2(aRow + k);
      const v2f a1 = ld2(aRow + k + 4);
      const v2f a2 = ld2(aRow + k + 8);
      const v2f a3 = ld2(aRow + k + 12);
      const v2f b0 = ld2(bRow + k + kr);
      const v2f b1 = ld2(bRow + k + kr + 4);
      const v2f b2 = ld2(bRow + k + kr + 8);
      const v2f b3 = ld2(bRow + k + kr + 12);
      acc = wmma_f32(a0, b0, acc);
      acc = wmma_f32(a1, b1, acc);
      acc = wmma_f32(a2, b2, acc);
      acc = wmma_f32(a3, b3, acc);
    }
    // K tail = {160,161} (+ zero-padded 162,163): attnLds cols >=162 are 0,
    // so every lane can safely load rows {160,161}.
    {
      const v2f at = ld2(aRow + 160);
      const v2f bt = ld2(bRow + 160);
      acc = wmma_f32(at, bt, acc);
    }
#pragma unroll
    for (int i = 0; i < 8; ++i) {
      const int M = mbase + i;
      if (M < P)
        out[((size_t)b * P + M) * C + c] = acc[i] * invN;
    }
  }
}

// ---------------------------------------------------------------------------
extern "C" void kernel_launch(void* const* d_in, const int* in_sizes, int n_in,
                              void* d_out, int out_size, void* d_ws, size_t ws_size,
                              hipStream_t stream) {
  const float* x    = (const float*)d_in[0];  // [B, C, H, W]
  const float* part = (const float*)d_in[1];  // [P, C]
  const float* pos  = (const float*)d_in[2];  // [N, C]
  float* out      = (float*)d_out;                       // [B, P*C] flat
  float* attn_out = out + (size_t)B * P * C;             // [B, P, N] flat
  float* posdot   = (float*)d_ws;                        // [16, 176] scratch

  posdot_kernel<<<NT, 256, 0, stream>>>(part, pos, posdot);
  safl_main_kernel<<<B, THREADS, 0, stream>>>(x, part, posdot, out, attn_out);
}